// FP_Layer_22531398435377
// MI455X (gfx1250) — compile-verified
//
#include <hip/hip_runtime.h>
#include <hip/hip_bf16.h>

typedef __attribute__((ext_vector_type(2))) float v2f;
typedef __attribute__((ext_vector_type(8))) float v8f;
typedef int vint4 __attribute__((vector_size(4 * sizeof(int))));

#ifndef __has_builtin
#define __has_builtin(x) 0
#endif

#if __has_builtin(__builtin_amdgcn_global_load_async_to_lds_b32) && \
    __has_builtin(__builtin_amdgcn_global_load_async_to_lds_b128)
#define USE_ASYNC_LDS 1
#else
#define USE_ASYNC_LDS 0
#endif

#define AS3I  __attribute__((address_space(3))) int*
#define AS3V4 __attribute__((address_space(3))) vint4*

#define B_     4
#define N_     16384
#define S_     2048
#define C1_    128
#define C2_    256
#define KCAT   (C1_ + C2_)   // 384
#define PTOT   (B_ * N_)     // 65536
#define H1     256
#define H2     128
#define CNT_F  65536.0f
#define BN_EPS_ 1e-5f

// ---------------- zero stats ----------------
__global__ void zero_stats_kernel(float* __restrict__ s, int n) {
  for (int i = threadIdx.x; i < n; i += 256) s[i] = 0.0f;
}

// ---------------- 3-NN + inverse-distance interpolation ----------------
// Writes interp rows [0,256) of Xcat (channel-major, 384 x PTOT).
__global__ __launch_bounds__(256)
void knn_interp_kernel(const float* __restrict__ xyz1,
                       const float* __restrict__ xyz2,
                       const float* __restrict__ feats2,
                       float* __restrict__ Xcat) {
  __shared__ float sx[S_], sy[S_], sz[S_];   // 24 KB of 320 KB WGP LDS
  const int b = blockIdx.y;
  const int n = blockIdx.x * 256 + threadIdx.x;
  const int p = b * N_ + n;

  for (int i = threadIdx.x; i < S_; i += 256) {
    const float* q = xyz2 + ((size_t)b * S_ + i) * 3;
    sx[i] = q[0]; sy[i] = q[1]; sz[i] = q[2];
  }
  __syncthreads();

  const float qx = xyz1[(size_t)p * 3 + 0];
  const float qy = xyz1[(size_t)p * 3 + 1];
  const float qz = xyz1[(size_t)p * 3 + 2];

  float d0 = 3.4e38f, d1 = 3.4e38f, d2 = 3.4e38f;
  int   i0 = 0, i1 = 0, i2 = 0;
  for (int s = 0; s < S_; ++s) {
    float dx = qx - sx[s], dy = qy - sy[s], dz = qz - sz[s];
    float d = dx * dx + dy * dy + dz * dz;
    if (d < d2) {
      if (d < d1) {
        d2 = d1; i2 = i1;
        if (d < d0) { d1 = d0; i1 = i0; d0 = d; i0 = s; }
        else        { d1 = d;  i1 = s; }
      } else { d2 = d; i2 = s; }
    }
  }

  float w0 = 1.0f / fmaxf(sqrtf(d0), 1e-8f);
  float w1 = 1.0f / fmaxf(sqrtf(d1), 1e-8f);
  float w2 = 1.0f / fmaxf(sqrtf(d2), 1e-8f);
  const float inv = 1.0f / (w0 + w1 + w2);
  w0 *= inv; w1 *= inv; w2 *= inv;

  const float* f2 = feats2 + (size_t)b * C2_ * S_;   // (256,2048) slab, L2-resident
  #pragma unroll 4
  for (int c = 0; c < C2_; ++c) {
    const float* row = f2 + (size_t)c * S_;
    Xcat[(size_t)c * PTOT + p] = w0 * row[i0] + w1 * row[i1] + w2 * row[i2];
  }
}

// ---------------- skip-feature copy into Xcat rows [256,384) ----------------
__global__ void copy_feats1_kernel(const float* __restrict__ f1, float* __restrict__ Xcat) {
  size_t t = (size_t)blockIdx.x * 256 + threadIdx.x;  // over B*C1*N
  int n = (int)(t % N_);
  int c = (int)((t / N_) % C1_);
  int b = (int)(t / ((size_t)N_ * C1_));
  Xcat[(size_t)(C2_ + c) * PTOT + (size_t)b * N_ + n] = f1[t];
}

// ---------------- WMMA f32 GEMM + fused BN-statistics ----------------
// Y[M][PTOT] = W[M][K] * X[K][PTOT]; also atomically accumulates per-row
// sum (stats[0..M)) and sum-of-squares (stats[M..2M)).
#define MT 64
#define NT 256
#define KC 16
#define ASTR 72    // pad: half-wave row offset 2*72 % 64 == 16 -> disjoint banks
#define BSTR 264   // pad: half-wave row offset 2*264 % 64 == 16 -> disjoint banks

__global__ __launch_bounds__(256)
void gemm_stats_kernel(const float* __restrict__ W,
                       const float* __restrict__ X,
                       float* __restrict__ Y,
                       float* __restrict__ stats,
                       int M, int K) {
  __shared__ float As[KC * ASTR];
  __shared__ float Bs[KC * BSTR];
  __shared__ float sstat[2 * MT];

  const int tid    = threadIdx.x;
  const int wid    = tid >> 5;          // 8 wave32s
  const int lane   = tid & 31;
  const int h      = lane >> 4;         // half-wave select (K pair ks+2h, ks+2h+1)
  const int ml     = lane & 15;         // M/N within fragment
  const int wave_m = wid & 3;           // 4 x 16 rows
  const int wave_n = wid >> 2;          // 2 x 128 cols
  const int m0 = blockIdx.y * MT;
  const int n0 = blockIdx.x * NT;
  const int am = wave_m * 16 + ml;

  if (tid < 2 * MT) sstat[tid] = 0.0f;

  v8f acc[8];
  #pragma unroll
  for (int j = 0; j < 8; ++j)
    #pragma unroll
    for (int e = 0; e < 8; ++e) acc[j][e] = 0.0f;

  for (int k0 = 0; k0 < K; k0 += KC) {
    __syncthreads();
#if USE_ASYNC_LDS
    // A tile (MT x KC): per-lane b32 async transfers, transpose folded into
    // the per-lane LDS destination address. Coalesced on K in global memory.
    #pragma unroll
    for (int r = 0; r < (MT * KC) / 256; ++r) {
      int idx = r * 256 + tid;
      int m = idx >> 4, kk = idx & 15;
      __builtin_amdgcn_global_load_async_to_lds_b32(
          (int*)(W + (size_t)(m0 + m) * K + (k0 + kk)),
          (AS3I)(As + kk * ASTR + m), 0, 0);
    }
    // B tile (KC x NT): 16B per lane, row-major, fully coalesced.
    #pragma unroll
    for (int r = 0; r < (KC * NT / 4) / 256; ++r) {
      int idx = r * 256 + tid;
      int row = idx >> 6, c4 = (idx & 63) << 2;
      __builtin_amdgcn_global_load_async_to_lds_b128(
          (vint4*)(X + (size_t)(k0 + row) * PTOT + n0 + c4),
          (AS3V4)(Bs + row * BSTR + c4), 0, 0);
    }
#if __has_builtin(__builtin_amdgcn_s_wait_asynccnt)
    __builtin_amdgcn_s_wait_asynccnt(0);
#else
    asm volatile("s_wait_asynccnt 0" ::: "memory");
#endif
#else
    // Fallback: synchronous staging. A transposed scalar, B as float4.
    #pragma unroll
    for (int r = 0; r < (MT * KC) / 256; ++r) {
      int idx = r * 256 + tid;
      int m = idx >> 4, kk = idx & 15;
      As[kk * ASTR + m] = W[(size_t)(m0 + m) * K + (k0 + kk)];
    }
    #pragma unroll
    for (int r = 0; r < (KC * NT / 4) / 256; ++r) {
      int idx = r * 256 + tid;
      int row = idx >> 6, c4 = (idx & 63) << 2;
      const float4 v = *(const float4*)(X + (size_t)(k0 + row) * PTOT + n0 + c4);
      *(float4*)(Bs + row * BSTR + c4) = v;
    }
#endif
    __syncthreads();

    #pragma unroll
    for (int ks = 0; ks < KC; ks += 4) {
      v2f a;
      a.x = As[(ks + 2 * h)     * ASTR + am];
      a.y = As[(ks + 2 * h + 1) * ASTR + am];
      #pragma unroll
      for (int j = 0; j < 8; ++j) {
        const int nn = wave_n * 128 + j * 16 + ml;
        v2f bf;
        bf.x = Bs[(ks + 2 * h)     * BSTR + nn];
        bf.y = Bs[(ks + 2 * h + 1) * BSTR + nn];
        acc[j] = __builtin_amdgcn_wmma_f32_16x16x4_f32(
            false, a, false, bf, (short)0, acc[j], false, false);
      }
    }
  }

  // Store Y and fold per-channel sum / sumsq (for training-mode BN).
  float rowsum[8], rowsq[8];
  #pragma unroll
  for (int v = 0; v < 8; ++v) { rowsum[v] = 0.0f; rowsq[v] = 0.0f; }

  #pragma unroll
  for (int j = 0; j < 8; ++j) {
    const int nn = n0 + wave_n * 128 + j * 16 + ml;
    #pragma unroll
    for (int v = 0; v < 8; ++v) {
      const float val = acc[j][v];
      const int mrow = m0 + wave_m * 16 + v + 8 * h;
      Y[(size_t)mrow * PTOT + nn] = val;
      rowsum[v] += val;
      rowsq[v]  += val * val;
    }
  }
  #pragma unroll
  for (int v = 0; v < 8; ++v) {
    #pragma unroll
    for (int off = 1; off < 16; off <<= 1) {
      rowsum[v] += __shfl_xor(rowsum[v], off, 32);
      rowsq[v]  += __shfl_xor(rowsq[v],  off, 32);
    }
  }
  if (ml == 0) {
    #pragma unroll
    for (int v = 0; v < 8; ++v) {
      const int rl = wave_m * 16 + v + 8 * h;
      atomicAdd(&sstat[rl], rowsum[v]);
      atomicAdd(&sstat[MT + rl], rowsq[v]);
    }
  }
  __syncthreads();
  if (tid < MT) {
    atomicAdd(&stats[m0 + tid],     sstat[tid]);
    atomicAdd(&stats[M + m0 + tid], sstat[MT + tid]);
  }
}

// ---------------- BN(train-mode, biased var) + ReLU, in place ----------------
__global__ void bn_relu_inplace_kernel(float* __restrict__ Y,
                                       const float* __restrict__ stats,
                                       const float* __restrict__ gamma,
                                       const float* __restrict__ beta, int M) {
  size_t t = (size_t)blockIdx.x * 256 + threadIdx.x;   // M * PTOT
  int c = (int)(t / PTOT);
  float mean = stats[c] * (1.0f / CNT_F);
  float var  = stats[M + c] * (1.0f / CNT_F) - mean * mean;
  float s = gamma[c] * rsqrtf(var + BN_EPS_);
  float sh = beta[c] - mean * s;
  Y[t] = fmaxf(Y[t] * s + sh, 0.0f);
}

// ---------------- BN + ReLU + scatter (o, b*n) -> (b, o, n) ----------------
__global__ void bn_relu_scatter_kernel(const float* __restrict__ Y,
                                       float* __restrict__ out,
                                       const float* __restrict__ stats,
                                       const float* __restrict__ gamma,
                                       const float* __restrict__ beta, int M) {
  size_t t = (size_t)blockIdx.x * 256 + threadIdx.x;   // M * PTOT
  int c = (int)(t / PTOT);
  size_t p = t % PTOT;
  int b = (int)(p / N_), n = (int)(p % N_);
  float mean = stats[c] * (1.0f / CNT_F);
  float var  = stats[M + c] * (1.0f / CNT_F) - mean * mean;
  float s = gamma[c] * rsqrtf(var + BN_EPS_);
  float sh = beta[c] - mean * s;
  float v = fmaxf(Y[t] * s + sh, 0.0f);
  out[((size_t)b * M + c) * N_ + n] = v;
}

extern "C" void kernel_launch(void* const* d_in, const int* in_sizes, int n_in,
                              void* d_out, int out_size, void* d_ws, size_t ws_size,
                              hipStream_t stream) {
  const float* xyz1   = (const float*)d_in[0];
  const float* xyz2   = (const float*)d_in[1];
  const float* feats1 = (const float*)d_in[2];
  const float* feats2 = (const float*)d_in[3];
  const float* W1     = (const float*)d_in[4];
  // d_in[5] (b1) and d_in[9] (b2) are mathematically absorbed by train-mode BN.
  const float* gamma1 = (const float*)d_in[6];
  const float* beta1  = (const float*)d_in[7];
  const float* W2     = (const float*)d_in[8];
  const float* gamma2 = (const float*)d_in[10];
  const float* beta2  = (const float*)d_in[11];
  float* out = (float*)d_out;

  float* ws     = (float*)d_ws;
  float* Xcat   = ws;                              // 384 x 65536
  float* Y1     = Xcat   + (size_t)KCAT * PTOT;    // 256 x 65536
  float* Y2     = Y1     + (size_t)H1   * PTOT;    // 128 x 65536
  float* stats1 = Y2     + (size_t)H2   * PTOT;    // 2*256
  float* stats2 = stats1 + 2 * H1;                 // 2*128

  zero_stats_kernel<<<1, 256, 0, stream>>>(stats1, 2 * (H1 + H2));

  knn_interp_kernel<<<dim3(N_ / 256, B_), 256, 0, stream>>>(xyz1, xyz2, feats2, Xcat);
  copy_feats1_kernel<<<(B_ * C1_ * N_) / 256, 256, 0, stream>>>(feats1, Xcat);

  gemm_stats_kernel<<<dim3(PTOT / NT, H1 / MT), 256, 0, stream>>>(W1, Xcat, Y1, stats1, H1, KCAT);
  bn_relu_inplace_kernel<<<(H1 * PTOT) / 256, 256, 0, stream>>>(Y1, stats1, gamma1, beta1, H1);

  gemm_stats_kernel<<<dim3(PTOT / NT, H2 / MT), 256, 0, stream>>>(W2, Y1, Y2, stats2, H2, H1);
  bn_relu_scatter_kernel<<<(H2 * PTOT) / 256, 256, 0, stream>>>(Y2, out, stats2, gamma2, beta2, H2);
}